// AttentionPositionalMixing_59785944760786
// MI455X (gfx1250) — compile-verified
//
#include <hip/hip_runtime.h>
#include <cmath>

typedef __attribute__((ext_vector_type(2))) float v2f;
typedef __attribute__((ext_vector_type(8))) float v8f;

#define WMMA_F32(a, b, c) \
  __builtin_amdgcn_wmma_f32_16x16x4_f32(false, (a), false, (b), (short)0, (c), false, false)

constexpr int Bn = 2, S = 1024, D = 512, H = 8, HD = 64;
constexpr int NROW = Bn * S;          // 2048
constexpr int CH = 64;                // chunk length
constexpr int NCHUNK = S / CH;        // 16
constexpr int STATE_STRIDE = HD * HD + HD; // 4160 floats per (bh,chunk) state

__device__ __forceinline__ float feature_map(float t) {
  // elu(t)+1 : t>0 ? t+1 : exp(t)
  return (t > 0.f) ? (t + 1.f) : __expf(t);
}

// ---------------------------------------------------------------------------
// Kernel 1: Q = fm(x@Wq), K = fm(x@Wk), V = x@Wv    [2048 x 512] x [512 x 512]
// block = 256 threads = 8 waves; blockIdx.x = row tile (16 rows);
// each wave owns one 64-col strip (4 WMMA accumulators). blockIdx.y = matrix.
// ---------------------------------------------------------------------------
__global__ __launch_bounds__(256) void k_qkv(
    const float* __restrict__ x,
    const float* __restrict__ Wq, const float* __restrict__ Wk,
    const float* __restrict__ Wv,
    float* __restrict__ Q, float* __restrict__ K, float* __restrict__ V) {
  const int which = blockIdx.y;
  const float* __restrict__ W = (which == 0) ? Wq : (which == 1) ? Wk : Wv;
  float* __restrict__ Out = (which == 0) ? Q : (which == 1) ? K : V;

  const int wave = threadIdx.x >> 5, lane = threadIdx.x & 31;
  const int hf = lane >> 4, l16 = lane & 15;
  const int row0 = blockIdx.x * 16;
  const int col0 = wave * 64;

  v8f acc[4] = {};
  for (int k = 0; k < D; k += 4) {
    const int kb = k + hf * 2;
    const float* ap = x + (size_t)(row0 + l16) * D + kb;
    v2f a; a.x = ap[0]; a.y = ap[1];
#pragma unroll
    for (int t = 0; t < 4; ++t) {
      const float* bp = W + (size_t)kb * D + col0 + t * 16 + l16;
      v2f bf; bf.x = bp[0]; bf.y = bp[D];
      acc[t] = WMMA_F32(a, bf, acc[t]);
    }
  }
#pragma unroll
  for (int t = 0; t < 4; ++t)
#pragma unroll
    for (int r = 0; r < 8; ++r) {
      const int row = row0 + hf * 8 + r;
      const int col = col0 + t * 16 + l16;
      float vv = acc[t][r];
      if (which < 2) vv = feature_map(vv);
      Out[(size_t)row * D + col] = vv;
    }
}

// ---------------------------------------------------------------------------
// Kernel 2: per (bh, chunk): state = K_c^T @ V_c  [64x64] and ksum = colsum(K_c)
// 256 blocks; 8 waves x 2 tiles = 16 WMMA tiles of the 64x64 state.
// ---------------------------------------------------------------------------
__global__ __launch_bounds__(256) void k_chunk_state(
    const float* __restrict__ K, const float* __restrict__ V,
    float* __restrict__ state) {
  const int bh = blockIdx.x >> 4, chunk = blockIdx.x & 15;
  const int b = bh >> 3, h = bh & 7;
  const int rowbase = b * S + chunk * CH;
  const int hc = h * HD;
  const int wave = threadIdx.x >> 5, lane = threadIdx.x & 31;
  const int hf = lane >> 4, l16 = lane & 15;
  float* __restrict__ st = state + (size_t)blockIdx.x * STATE_STRIDE;

#pragma unroll
  for (int tt = 0; tt < 2; ++tt) {
    const int t = wave * 2 + tt;
    const int m0 = (t >> 2) * 16, n0 = (t & 3) * 16;
    v8f acc = {};
    for (int kk = 0; kk < CH; kk += 4) {
      const int kb = kk + hf * 2;
      // A = K^T : A[m][k] = K[s0+k][hc+m]   (coalesced over m)
      const float* ka = K + (size_t)(rowbase + kb) * D + hc + m0 + l16;
      v2f a; a.x = ka[0]; a.y = ka[D];
      // B = V  : B[k][n] = V[s0+k][hc+n]    (coalesced over n)
      const float* vp = V + (size_t)(rowbase + kb) * D + hc + n0 + l16;
      v2f bf; bf.x = vp[0]; bf.y = vp[D];
      acc = WMMA_F32(a, bf, acc);
    }
#pragma unroll
    for (int r = 0; r < 8; ++r)
      st[(m0 + hf * 8 + r) * HD + n0 + l16] = acc[r];
  }
  if (threadIdx.x < HD) {
    float s = 0.f;
    for (int ss = 0; ss < CH; ++ss)
      s += K[(size_t)(rowbase + ss) * D + hc + threadIdx.x];
    st[HD * HD + threadIdx.x] = s;
  }
}

// ---------------------------------------------------------------------------
// Kernel 3: per (bh, chunk): sum predecessor states -> S_prev, ksum_prev;
//   Aqk = (Q_c @ K_c^T) * causal mask; den = rowsum(Aqk) + Q.ksum_prev + 1e-6;
//   att = (Q_c @ S_prev + Aqk @ V_c) / den
// ---------------------------------------------------------------------------
__global__ __launch_bounds__(256) void k_attn(
    const float* __restrict__ Q, const float* __restrict__ K,
    const float* __restrict__ V, const float* __restrict__ state,
    float* __restrict__ att) {
  __shared__ float Sprev[HD * HD];
  __shared__ float Amask[CH * CH];
  __shared__ float ksum[HD];
  __shared__ float den[CH];

  const int bh = blockIdx.x >> 4, chunk = blockIdx.x & 15;
  const int b = bh >> 3, h = bh & 7;
  const int rowbase = b * S + chunk * CH;
  const int hc = h * HD;
  const int wave = threadIdx.x >> 5, lane = threadIdx.x & 31;
  const int hf = lane >> 4, l16 = lane & 15;

  // 1) exclusive prefix over previous chunk states (4096+64 floats)
  {
    float accv[16];
#pragma unroll
    for (int i = 0; i < 16; ++i) accv[i] = 0.f;
    float ks = 0.f;
    for (int c = 0; c < chunk; ++c) {
      const float* st = state + (size_t)(bh * NCHUNK + c) * STATE_STRIDE;
#pragma unroll
      for (int i = 0; i < 16; ++i) accv[i] += st[threadIdx.x * 16 + i];
      if (threadIdx.x < HD) ks += st[HD * HD + threadIdx.x];
    }
#pragma unroll
    for (int i = 0; i < 16; ++i) Sprev[threadIdx.x * 16 + i] = accv[i];
    if (threadIdx.x < HD) ksum[threadIdx.x] = ks;
  }
  __syncthreads();

  // 2) Aqk = Q_c @ K_c^T, causal-masked (j <= i)
#pragma unroll
  for (int tt = 0; tt < 2; ++tt) {
    const int t = wave * 2 + tt;
    const int m0 = (t >> 2) * 16, n0 = (t & 3) * 16;
    v8f acc = {};
    for (int kk = 0; kk < HD; kk += 4) {
      const int kb = kk + hf * 2;
      const float* qa = Q + (size_t)(rowbase + m0 + l16) * D + hc + kb;
      v2f a; a.x = qa[0]; a.y = qa[1];
      // B = K^T : B[k][n] = K[s0+n][hc+k]
      const float* kp = K + (size_t)(rowbase + n0 + l16) * D + hc + kb;
      v2f bf; bf.x = kp[0]; bf.y = kp[1];
      acc = WMMA_F32(a, bf, acc);
    }
#pragma unroll
    for (int r = 0; r < 8; ++r) {
      const int i = m0 + hf * 8 + r;
      const int j = n0 + l16;
      Amask[i * CH + j] = (j <= i) ? acc[r] : 0.f;
    }
  }
  __syncthreads();

  // 3) denominator per row
  if (threadIdx.x < CH) {
    const int i = threadIdx.x;
    float s = 0.f;
    for (int j = 0; j <= i; ++j) s += Amask[i * CH + j];
    const float* qrow = Q + (size_t)(rowbase + i) * D + hc;
    float s2 = 0.f;
    for (int d0 = 0; d0 < HD; ++d0) s2 += qrow[d0] * ksum[d0];
    den[i] = s + s2 + 1e-6f;
  }
  __syncthreads();

  // 4) num = Q_c @ S_prev + Amask @ V_c ; att = num / den
#pragma unroll
  for (int tt = 0; tt < 2; ++tt) {
    const int t = wave * 2 + tt;
    const int m0 = (t >> 2) * 16, n0 = (t & 3) * 16;
    v8f acc = {};
    for (int kk = 0; kk < HD; kk += 4) {
      const int kb = kk + hf * 2;
      // inter-chunk: Q @ S_prev
      const float* qa = Q + (size_t)(rowbase + m0 + l16) * D + hc + kb;
      v2f a; a.x = qa[0]; a.y = qa[1];
      v2f bf; bf.x = Sprev[kb * HD + n0 + l16];
      bf.y = Sprev[(kb + 1) * HD + n0 + l16];
      acc = WMMA_F32(a, bf, acc);
      // intra-chunk: Amask @ V
      v2f a2; a2.x = Amask[(m0 + l16) * CH + kb];
      a2.y = Amask[(m0 + l16) * CH + kb + 1];
      const float* vp = V + (size_t)(rowbase + kb) * D + hc + n0 + l16;
      v2f b2; b2.x = vp[0]; b2.y = vp[D];
      acc = WMMA_F32(a2, b2, acc);
    }
#pragma unroll
    for (int r = 0; r < 8; ++r) {
      const int i = m0 + hf * 8 + r;
      const int n = n0 + l16;
      att[(size_t)(rowbase + i) * D + hc + n] = acc[r] / den[i];
    }
  }
}

// ---------------------------------------------------------------------------
// Kernel 4: out = att @ Wo + bo
// ---------------------------------------------------------------------------
__global__ __launch_bounds__(256) void k_outproj(
    const float* __restrict__ att, const float* __restrict__ Wo,
    const float* __restrict__ bo, float* __restrict__ out) {
  const int wave = threadIdx.x >> 5, lane = threadIdx.x & 31;
  const int hf = lane >> 4, l16 = lane & 15;
  const int row0 = blockIdx.x * 16;
  const int col0 = wave * 64;

  v8f acc[4] = {};
  for (int k = 0; k < D; k += 4) {
    const int kb = k + hf * 2;
    const float* ap = att + (size_t)(row0 + l16) * D + kb;
    v2f a; a.x = ap[0]; a.y = ap[1];
#pragma unroll
    for (int t = 0; t < 4; ++t) {
      const float* bp = Wo + (size_t)kb * D + col0 + t * 16 + l16;
      v2f bf; bf.x = bp[0]; bf.y = bp[D];
      acc[t] = WMMA_F32(a, bf, acc[t]);
    }
  }
#pragma unroll
  for (int t = 0; t < 4; ++t)
#pragma unroll
    for (int r = 0; r < 8; ++r) {
      const int row = row0 + hf * 8 + r;
      const int col = col0 + t * 16 + l16;
      out[(size_t)row * D + col] = acc[t][r] + bo[col];
    }
}

// ---------------------------------------------------------------------------
// Kernel 5: g = sigmoid([x | out] @ Wg + bg) ; y = x + g*(out - x)
// K = 1024: first 512 k-values read x, next 512 read out.
// ---------------------------------------------------------------------------
__global__ __launch_bounds__(256) void k_gate(
    const float* __restrict__ x, const float* __restrict__ out,
    const float* __restrict__ Wg, const float* __restrict__ bg,
    float* __restrict__ y) {
  const int wave = threadIdx.x >> 5, lane = threadIdx.x & 31;
  const int hf = lane >> 4, l16 = lane & 15;
  const int row0 = blockIdx.x * 16;
  const int col0 = wave * 64;

  v8f acc[4] = {};
  for (int k = 0; k < 2 * D; k += 4) {
    const int kb = k + hf * 2;  // kb, kb+1 stay on same side of the 512 split
    const float* asrc = (kb < D) ? (x + (size_t)(row0 + l16) * D + kb)
                                 : (out + (size_t)(row0 + l16) * D + (kb - D));
    v2f a; a.x = asrc[0]; a.y = asrc[1];
#pragma unroll
    for (int t = 0; t < 4; ++t) {
      const float* bp = Wg + (size_t)kb * D + col0 + t * 16 + l16;
      v2f bf; bf.x = bp[0]; bf.y = bp[D];
      acc[t] = WMMA_F32(a, bf, acc[t]);
    }
  }
#pragma unroll
  for (int t = 0; t < 4; ++t)
#pragma unroll
    for (int r = 0; r < 8; ++r) {
      const int row = row0 + hf * 8 + r;
      const int col = col0 + t * 16 + l16;
      const float g = 1.f / (1.f + __expf(-(acc[t][r] + bg[col])));
      const float xv = x[(size_t)row * D + col];
      const float ov = out[(size_t)row * D + col];
      y[(size_t)row * D + col] = xv + g * (ov - xv);
    }
}

// ---------------------------------------------------------------------------
extern "C" void kernel_launch(void* const* d_in, const int* in_sizes, int n_in,
                              void* d_out, int out_size, void* d_ws,
                              size_t ws_size, hipStream_t stream) {
  const float* x  = (const float*)d_in[0];
  const float* Wq = (const float*)d_in[1];
  const float* Wk = (const float*)d_in[2];
  const float* Wv = (const float*)d_in[3];
  const float* Wo = (const float*)d_in[4];
  const float* bo = (const float*)d_in[5];
  const float* Wg = (const float*)d_in[6];
  const float* bg = (const float*)d_in[7];
  float* y = (float*)d_out;

  float* ws = (float*)d_ws;
  const size_t ND = (size_t)NROW * D;
  float* Q     = ws;
  float* K     = Q + ND;
  float* V     = K + ND;
  float* ATT   = V + ND;
  float* OUT   = ATT + ND;
  float* STATE = OUT + ND;  // 256 * 4160 floats

  k_qkv<<<dim3(NROW / 16, 3), 256, 0, stream>>>(x, Wq, Wk, Wv, Q, K, V);
  k_chunk_state<<<Bn * H * NCHUNK, 256, 0, stream>>>(K, V, STATE);
  k_attn<<<Bn * H * NCHUNK, 256, 0, stream>>>(Q, K, V, STATE, ATT);
  k_outproj<<<NROW / 16, 256, 0, stream>>>(ATT, Wo, bo, OUT);
  k_gate<<<NROW / 16, 256, 0, stream>>>(x, OUT, Wg, bg, y);
}